// BicameralAttention_18769007084059
// MI455X (gfx1250) — compile-verified
//
#include <hip/hip_runtime.h>
#include <hip/hip_bf16.h>

typedef float v2f __attribute__((ext_vector_type(2)));
typedef float v8f __attribute__((ext_vector_type(8)));

#define BB   2
#define LL   2048
#define DD   1024
#define HH   16
#define HDIM 64
#define MTOT (BB*LL)   // 4096

__device__ __forceinline__ v8f wmma_f32(v2f a, v2f b, v8f c) {
  // D = A(16x4) * B(4x16) + C(16x16), fp32, wave32
  return __builtin_amdgcn_wmma_f32_16x16x4_f32(
      /*neg_a=*/false, a, /*neg_b=*/false, b,
      /*c_mod=*/(short)0, c, /*reuse_a=*/false, /*reuse_b=*/false);
}

// Async copy 16 bytes global -> LDS (per lane), tracked by ASYNCcnt.
// GVS mode: addr = SGPR64 base + VGPR32 byte offset.
__device__ __forceinline__ void async_ld16(unsigned lds_addr,
                                           const float* base, unsigned off) {
  asm volatile("global_load_async_to_lds_b128 %0, %1, %2"
               :: "v"(lds_addr), "v"(off), "s"(base) : "memory");
}
__device__ __forceinline__ void wait_async0() {
  asm volatile("s_wait_asynccnt 0" ::: "memory");
}
__device__ __forceinline__ unsigned lds_addr_of(const void* p) {
  return (unsigned)(uintptr_t)p;   // low 32 bits of flat shared ptr = LDS offset
}

// ---------------------------------------------------------------------------
// GEMM: Y[M=4096, N=1024] = X[4096,1024] @ W[1024,1024] + bias
// WG = 4 waves, 64x64 block. Double-buffered async-to-LDS staging,
// one barrier per 16-wide K chunk; fp32 WMMA 16x16x4.
// ---------------------------------------------------------------------------
__global__ __launch_bounds__(128) void bicam_proj_wmma(
    const float* __restrict__ X, const float* __restrict__ W,
    const float* __restrict__ bias, float* __restrict__ Y)
{
  __shared__ __align__(16) float lx[2][64][20];  // 64 rows x 16 k (+4 pad)
  __shared__ __align__(16) float lw[2][16][68];  // 16 k x 64 cols (+4 pad)

  const int tid  = threadIdx.x;
  const int lane = tid & 31, wave = tid >> 5;
  const int hl   = lane >> 4;     // lane-half (K split per ISA A/B layout)
  const int l16  = lane & 15;
  const int row0 = blockIdx.y * 64;
  const int col0 = blockIdx.x * 64;

  // staging assignment: 2 chunks of 16B per thread for each of X and W tiles
  unsigned ldsX[2][2], ldsW[2][2], offX[2], offW[2];
  #pragma unroll
  for (int i = 0; i < 2; ++i) {
    int idx = tid + i * 128;
    int rX = idx >> 2, cX = (idx & 3) * 4;    // X tile 64x16
    int rW = idx >> 4, cW = (idx & 15) * 4;   // W tile 16x64
    offX[i] = (unsigned)(((row0 + rX) * DD + cX) * 4);       // + kk0*4
    offW[i] = (unsigned)((rW * DD + col0 + cW) * 4);         // + kk0*4096
    #pragma unroll
    for (int bf = 0; bf < 2; ++bf) {
      ldsX[bf][i] = lds_addr_of(&lx[bf][rX][cX]);
      ldsW[bf][i] = lds_addr_of(&lw[bf][rW][cW]);
    }
  }

  v8f acc[4];
  for (int nt = 0; nt < 4; ++nt)
    for (int g = 0; g < 8; ++g) acc[nt][g] = 0.0f;

  // prefetch first K-chunk into buffer 0
  #pragma unroll
  for (int i = 0; i < 2; ++i) {
    async_ld16(ldsX[0][i], X, offX[i]);
    async_ld16(ldsW[0][i], W, offW[i]);
  }

  for (int kk0 = 0; kk0 < DD; kk0 += 16) {
    const int buf = (kk0 >> 4) & 1;
    wait_async0();        // this wave's chunks landed
    __syncthreads();      // everyone's chunks landed; other buffer free
    if (kk0 + 16 < DD) {
      #pragma unroll
      for (int i = 0; i < 2; ++i) {
        async_ld16(ldsX[buf ^ 1][i], X, offX[i] + (unsigned)(kk0 + 16) * 4u);
        async_ld16(ldsW[buf ^ 1][i], W, offW[i] + (unsigned)(kk0 + 16) * 4096u);
      }
    }
    #pragma unroll
    for (int kk = 0; kk < 4; ++kk) {
      v2f a;  // A 16x4: lane l16 = M, VGPR0 holds K = hl*2, VGPR1 K = hl*2+1
      a.x = lx[buf][wave * 16 + l16][kk * 4 + hl * 2 + 0];
      a.y = lx[buf][wave * 16 + l16][kk * 4 + hl * 2 + 1];
      #pragma unroll
      for (int nt = 0; nt < 4; ++nt) {
        v2f b;  // B 4x16: lane l16 = N
        b.x = lw[buf][kk * 4 + hl * 2 + 0][nt * 16 + l16];
        b.y = lw[buf][kk * 4 + hl * 2 + 1][nt * 16 + l16];
        acc[nt] = wmma_f32(a, b, acc[nt]);
      }
    }
  }

  #pragma unroll
  for (int nt = 0; nt < 4; ++nt) {
    int col = col0 + nt * 16 + l16;
    float bv = bias[col];
    #pragma unroll
    for (int g = 0; g < 8; ++g) {
      int row = row0 + wave * 16 + hl * 8 + g;   // C layout: row = half*8 + g
      Y[(size_t)row * DD + col] = acc[nt][g] + bv;
    }
  }
}

// ---------------------------------------------------------------------------
// Attention for one (b,h), 64 q-rows per WG (16 per wave).
// Pass 1: online row-max/row-sum.  Pass 2: write normalized P + O = P@V.
// K/V tiles double-buffered via async-to-LDS; one barrier per tile.
// ---------------------------------------------------------------------------
__global__ __launch_bounds__(128) void bicam_attn_wmma(
    const float* __restrict__ Qw, const float* __restrict__ Kw,
    const float* __restrict__ Vw, float* __restrict__ attn_l,
    float* __restrict__ attn_g, float* __restrict__ Oh,
    const float* __restrict__ lsc, const float* __restrict__ gsc)
{
  __shared__ __align__(16) float lk[2][16][68];
  __shared__ __align__(16) float lv[2][16][68];
  __shared__ __align__(16) float lp[4][16][17];

  const int tid  = threadIdx.x;
  const int lane = tid & 31, wave = tid >> 5;
  const int hl   = lane >> 4, l16 = lane & 15;
  const int bh   = blockIdx.y;
  const int b    = bh >> 4, h = bh & 15;
  const bool causal = (h < 8);
  const int q0b = blockIdx.x * 64;
  const int q0w = q0b + wave * 16;

  const float scale = 0.125f * (causal ? lsc[0] : gsc[0]);  // 1/sqrt(64)=0.125
  const float* Qb = Qw + (size_t)b * LL * DD + (size_t)h * HDIM;
  const float* Kb = Kw + (size_t)b * LL * DD + (size_t)h * HDIM;
  const float* Vb = Vw + (size_t)b * LL * DD + (size_t)h * HDIM;
  float* Pout = causal ? (attn_l + (size_t)(b * 8 + h)       * LL * LL)
                       : (attn_g + (size_t)(b * 8 + (h - 8)) * LL * LL);

  const int kend_blk = causal ? (q0b + 64) : LL;  // block-uniform loop bound
  const int kend_wav = causal ? (q0w + 16) : LL;  // this wave's compute bound

  // staging: 2 chunks of 16B per thread per tile (16 rows x 64 dims)
  unsigned ldsK[2][2], ldsV[2][2], offKV[2];
  #pragma unroll
  for (int i = 0; i < 2; ++i) {
    int idx = tid + i * 128;
    int r = idx >> 4, c = (idx & 15) * 4;
    offKV[i] = (unsigned)((r * DD + c) * 4);      // + k0*DD*4
    #pragma unroll
    for (int bf = 0; bf < 2; ++bf) {
      ldsK[bf][i] = lds_addr_of(&lk[bf][r][c]);
      ldsV[bf][i] = lds_addr_of(&lv[bf][r][c]);
    }
  }

  // Q fragments (A matrix, 16 rows x 64 dims = 16 chunks of K=4), kept live.
  v2f aq[16];
  {
    const float* qr = Qb + (size_t)(q0w + l16) * DD;
    #pragma unroll
    for (int kd = 0; kd < 16; ++kd) {
      aq[kd].x = qr[kd * 4 + hl * 2 + 0];
      aq[kd].y = qr[kd * 4 + hl * 2 + 1];
    }
  }

  float m[8], lse[8];
  #pragma unroll
  for (int g = 0; g < 8; ++g) { m[g] = -3.0e38f; lse[g] = 0.0f; }

  // ---------------- pass 1: softmax statistics ----------------
  #pragma unroll
  for (int i = 0; i < 2; ++i) async_ld16(ldsK[0][i], Kb, offKV[i]);

  for (int k0 = 0; k0 < kend_blk; k0 += 16) {
    const int buf = (k0 >> 4) & 1;
    wait_async0();
    __syncthreads();
    if (k0 + 16 < kend_blk) {
      #pragma unroll
      for (int i = 0; i < 2; ++i)
        async_ld16(ldsK[buf ^ 1][i], Kb, offKV[i] + (unsigned)(k0 + 16) * (DD * 4u));
    }
    if (k0 < kend_wav) {
      v8f s; for (int g = 0; g < 8; ++g) s[g] = 0.0f;
      #pragma unroll
      for (int kd = 0; kd < 16; ++kd) {
        v2f bk;  // B = K^T tile: B(d, n) = K[k0+n][d]
        bk.x = lk[buf][l16][kd * 4 + hl * 2 + 0];
        bk.y = lk[buf][l16][kd * 4 + hl * 2 + 1];
        s = wmma_f32(aq[kd], bk, s);
      }
      #pragma unroll
      for (int g = 0; g < 8; ++g) {
        float val = s[g] * scale;
        if (causal && (k0 + l16 > q0w + hl * 8 + g)) val = -3.0e38f;
        float tm = val;   // row-reduce across the 16 lanes of this half
        tm = fmaxf(tm, __shfl_xor(tm, 1, 32));
        tm = fmaxf(tm, __shfl_xor(tm, 2, 32));
        tm = fmaxf(tm, __shfl_xor(tm, 4, 32));
        tm = fmaxf(tm, __shfl_xor(tm, 8, 32));
        float mn = fmaxf(m[g], tm);
        float p  = __expf(val - mn);
        float ps = p;
        ps += __shfl_xor(ps, 1, 32);
        ps += __shfl_xor(ps, 2, 32);
        ps += __shfl_xor(ps, 4, 32);
        ps += __shfl_xor(ps, 8, 32);
        lse[g] = lse[g] * __expf(m[g] - mn) + ps;
        m[g]   = mn;
      }
    }
  }

  float rinv[8];
  #pragma unroll
  for (int g = 0; g < 8; ++g) rinv[g] = 1.0f / lse[g];

  // ---------------- pass 2: write P, O = P @ V ----------------
  v8f o[4];
  for (int nt = 0; nt < 4; ++nt)
    for (int g = 0; g < 8; ++g) o[nt][g] = 0.0f;

  __syncthreads();   // pass-1 reads fully done before pass-2 prefetch lands
  #pragma unroll
  for (int i = 0; i < 2; ++i) {
    async_ld16(ldsK[0][i], Kb, offKV[i]);
    async_ld16(ldsV[0][i], Vb, offKV[i]);
  }

  for (int k0 = 0; k0 < kend_blk; k0 += 16) {
    const int buf = (k0 >> 4) & 1;
    wait_async0();
    __syncthreads();
    if (k0 + 16 < kend_blk) {
      #pragma unroll
      for (int i = 0; i < 2; ++i) {
        unsigned o16 = offKV[i] + (unsigned)(k0 + 16) * (DD * 4u);
        async_ld16(ldsK[buf ^ 1][i], Kb, o16);
        async_ld16(ldsV[buf ^ 1][i], Vb, o16);
      }
    }
    if (k0 < kend_wav) {
      v8f s; for (int g = 0; g < 8; ++g) s[g] = 0.0f;
      #pragma unroll
      for (int kd = 0; kd < 16; ++kd) {
        v2f bk;
        bk.x = lk[buf][l16][kd * 4 + hl * 2 + 0];
        bk.y = lk[buf][l16][kd * 4 + hl * 2 + 1];
        s = wmma_f32(aq[kd], bk, s);
      }
      #pragma unroll
      for (int g = 0; g < 8; ++g) {
        int row = hl * 8 + g;
        bool ok = !causal || (k0 + l16 <= q0w + row);
        float p = ok ? __expf(s[g] * scale - m[g]) * rinv[g] : 0.0f;
        lp[wave][row][l16] = p;                                  // C->A bounce
        Pout[(size_t)(q0w + row) * LL + (k0 + l16)] = p;         // materialize
      }
      #pragma unroll
      for (int kd = 0; kd < 4; ++kd) {
        v2f ap;  // A = P: m = l16, k = key index
        ap.x = lp[wave][l16][kd * 4 + hl * 2 + 0];
        ap.y = lp[wave][l16][kd * 4 + hl * 2 + 1];
        #pragma unroll
        for (int nt = 0; nt < 4; ++nt) {
          v2f bv;  // B = V: B(k, n) = V[key][nt*16+n]
          bv.x = lv[buf][kd * 4 + hl * 2 + 0][nt * 16 + l16];
          bv.y = lv[buf][kd * 4 + hl * 2 + 1][nt * 16 + l16];
          o[nt] = wmma_f32(ap, bv, o[nt]);
        }
      }
    }
  }

  // causal: zero the strictly-above-diagonal remainder (softmax of -1e30 == 0)
  if (causal && (q0w + 16 < LL)) {
    const int ks  = q0w + 16;
    const int cnt = LL - ks;
    for (int r = 0; r < 16; ++r) {
      float* rp = Pout + (size_t)(q0w + r) * LL + ks;
      for (int c = lane * 4; c < cnt; c += 128)
        *(float4*)(rp + c) = make_float4(0.f, 0.f, 0.f, 0.f);
    }
  }

  // write O tile into [B*L, D] head-concat layout for the final projection
  float* Ob = Oh + (size_t)b * LL * DD + (size_t)h * HDIM;
  #pragma unroll
  for (int nt = 0; nt < 4; ++nt)
    #pragma unroll
    for (int g = 0; g < 8; ++g)
      Ob[(size_t)(q0w + hl * 8 + g) * DD + nt * 16 + l16] = o[nt][g];
}

// ---------------------------------------------------------------------------
extern "C" void kernel_launch(void* const* d_in, const int* in_sizes, int n_in,
                              void* d_out, int out_size, void* d_ws, size_t ws_size,
                              hipStream_t stream) {
  const float* x   = (const float*)d_in[0];
  const float* Wq  = (const float*)d_in[1];
  const float* bq  = (const float*)d_in[2];
  const float* Wk  = (const float*)d_in[3];
  const float* bk  = (const float*)d_in[4];
  const float* Wv  = (const float*)d_in[5];
  const float* bv  = (const float*)d_in[6];
  const float* Wo  = (const float*)d_in[7];
  const float* bo  = (const float*)d_in[8];
  const float* lsc = (const float*)d_in[9];
  const float* gsc = (const float*)d_in[10];

  float* out    = (float*)d_out;                       // [B,L,D]
  float* attn_l = out + (size_t)BB * LL * DD;          // [B,8,L,L]
  float* attn_g = attn_l + (size_t)BB * 8 * LL * LL;   // [B,8,L,L]

  float* ws = (float*)d_ws;                            // 16M floats = 64 MB
  float* Q  = ws;
  float* K  = Q + (size_t)BB * LL * DD;
  float* V  = K + (size_t)BB * LL * DD;
  float* Oh = V + (size_t)BB * LL * DD;

  dim3 pg(DD / 64, MTOT / 64);     // (16, 64)
  bicam_proj_wmma<<<pg, 128, 0, stream>>>(x, Wq, bq, Q);
  bicam_proj_wmma<<<pg, 128, 0, stream>>>(x, Wk, bk, K);
  bicam_proj_wmma<<<pg, 128, 0, stream>>>(x, Wv, bv, V);

  dim3 ag(LL / 64, BB * HH);       // (32, 32)
  bicam_attn_wmma<<<ag, 128, 0, stream>>>(Q, K, V, attn_l, attn_g, Oh, lsc, gsc);

  bicam_proj_wmma<<<pg, 128, 0, stream>>>(Oh, Wo, bo, out);
}